// HashCL_53197464928383
// MI455X (gfx1250) — compile-verified
//
#include <hip/hip_runtime.h>
#include <hip/hip_bf16.h>

#define NUM_USERS 60000
#define NUM_ITEMS 40000
#define N_NODES   (NUM_USERS + NUM_ITEMS)
#define N_EDGES   2000000
#define CON_DIM   64
#define BIN_DIM   64
#define OUT_ROW   128            // N_LAYERS * 64
#define CL_EPS    0.2f

typedef __attribute__((ext_vector_type(2))) float v2f;
typedef __attribute__((ext_vector_type(8))) float v8f;

__device__ __forceinline__ float fsign(float x) {
    return (x > 0.f) ? 1.f : ((x < 0.f) ? -1.f : 0.f);
}

// -------------------------------------------------------------------------
// Kernel 1: zero the con slice for layer l (atomic accumulation target).
// -------------------------------------------------------------------------
__global__ void zero_slice_kernel(float* __restrict__ conout, int l) {
    int idx = blockIdx.x * blockDim.x + threadIdx.x;     // over N_NODES*64
    if (idx >= N_NODES * CON_DIM) return;
    int node = idx >> 6;
    int d    = idx & 63;
    conout[node * OUT_ROW + l * CON_DIM + d] = 0.f;
}

// -------------------------------------------------------------------------
// Kernel 2: SpMM scatter  y[r,d] += vals[e] * x[cols[e], d]
// One thread per (edge, dim): the 64 threads of an edge read a contiguous
// 256B row (coalesced, L2-resident) and issue native f32 atomics.
// Layer 0 gathers from the two input embedding tables; layer l>0 gathers
// from the previous layer's (post-noise) con slice living in d_out.
// -------------------------------------------------------------------------
__global__ void spmm_scatter_kernel(const int* __restrict__ rows,
                                    const int* __restrict__ cols,
                                    const float* __restrict__ vals,
                                    const float* __restrict__ user,
                                    const float* __restrict__ item,
                                    float* __restrict__ conout, int l) {
    int gid = blockIdx.x * blockDim.x + threadIdx.x;     // over N_EDGES*64
    if (gid >= N_EDGES * 64) return;
    int e = gid >> 6;
    int d = gid & 63;
    int r = rows[e];
    int c = cols[e];
    float v = vals[e];
    float x;
    if (l == 0) {
        x = (c < NUM_USERS) ? user[c * CON_DIM + d]
                            : item[(c - NUM_USERS) * CON_DIM + d];
    } else {
        x = conout[c * OUT_ROW + (l - 1) * CON_DIM + d];
    }
    unsafeAtomicAdd(&conout[r * OUT_ROW + l * CON_DIM + d], v * x);
    // native global_atomic_add_f32 (L2-resident scatter)
}

// -------------------------------------------------------------------------
// Kernel 3: hash = sign(y @ W^T) via V_WMMA_F32_16X16X4_F32 (exact f32).
// One wave per 16-node tile. h[n,b] = sum_d y[n,d]*W[b,d]:
//   A (16x4, M=node, K=d):  lane m=L%16 holds row M=m; K = kb + 2*(L/16)+{0,1}
//   B (4x16, K=d, N=b):     lane m holds col N=m (bin idx t*16+m); same K split
// Both operands are consecutive-K float2 loads -> simple vector loads.
// 4 accumulators (BIN_DIM=64 -> 4 tiles of N=16), 16 K-steps of 4 WMMAs.
// -------------------------------------------------------------------------
__global__ void hash_wmma_kernel(const float* __restrict__ conbase,
                                 const float* __restrict__ W,
                                 float* __restrict__ binbase, int l) {
    int wave = threadIdx.x >> 5;              // 8 waves / 256-thread block
    int lane = threadIdx.x & 31;
    int tile = blockIdx.x * 8 + wave;
    int node_base = tile * 16;
    if (node_base >= N_NODES) return;         // uniform per wave: EXEC all-1s

    int m  = lane & 15;
    int hi = lane >> 4;

    const float* yrow = conbase + (node_base + m) * OUT_ROW + l * CON_DIM + 2 * hi;
    const float* w0 = W + ( 0 + m) * CON_DIM + 2 * hi;
    const float* w1 = W + (16 + m) * CON_DIM + 2 * hi;
    const float* w2 = W + (32 + m) * CON_DIM + 2 * hi;
    const float* w3 = W + (48 + m) * CON_DIM + 2 * hi;

    v8f c0 = {}, c1 = {}, c2 = {}, c3 = {};
#pragma unroll
    for (int kb = 0; kb < CON_DIM; kb += 4) {
        v2f a  = *(const v2f*)(yrow + kb);
        v2f b0 = *(const v2f*)(w0 + kb);
        v2f b1 = *(const v2f*)(w1 + kb);
        v2f b2 = *(const v2f*)(w2 + kb);
        v2f b3 = *(const v2f*)(w3 + kb);
        c0 = __builtin_amdgcn_wmma_f32_16x16x4_f32(false, a, false, b0, (short)0, c0, false, false);
        c1 = __builtin_amdgcn_wmma_f32_16x16x4_f32(false, a, false, b1, (short)0, c1, false, false);
        c2 = __builtin_amdgcn_wmma_f32_16x16x4_f32(false, a, false, b2, (short)0, c2, false, false);
        c3 = __builtin_amdgcn_wmma_f32_16x16x4_f32(false, a, false, b3, (short)0, c3, false, false);
    }

    // C/D layout: lane L -> N = L%16; VGPR v -> M = v + 8*(L/16)
#pragma unroll
    for (int v = 0; v < 8; ++v) {
        int node = node_base + v + 8 * hi;
        int base = node * OUT_ROW + l * BIN_DIM + m;
        binbase[base +  0] = fsign(c0[v]);
        binbase[base + 16] = fsign(c1[v]);
        binbase[base + 32] = fsign(c2[v]);
        binbase[base + 48] = fsign(c3[v]);
    }
}

// -------------------------------------------------------------------------
// Kernel 4: noise injection  y += sign(y) * (noise_row/max(||row||,eps)) * 0.2
// One wave (32 lanes) per node, 2 dims/lane, wave32 shuffle reduction.
// In-place on the con slice (which then feeds the next layer's SpMM).
// -------------------------------------------------------------------------
__global__ void noise_update_kernel(const float* __restrict__ noise,
                                    float* __restrict__ conbase, int l) {
    int wave = threadIdx.x >> 5;
    int lane = threadIdx.x & 31;
    int node = blockIdx.x * 8 + wave;
    if (node >= N_NODES) return;

    const float* nrow = noise + l * (N_NODES * CON_DIM) + node * CON_DIM;
    v2f nv = *(const v2f*)(nrow + 2 * lane);
    float ss = nv.x * nv.x + nv.y * nv.y;
#pragma unroll
    for (int off = 16; off > 0; off >>= 1)
        ss += __shfl_xor(ss, off, 32);
    float scale = CL_EPS / fmaxf(sqrtf(ss), 1e-12f);

    float* yp = conbase + node * OUT_ROW + l * CON_DIM + 2 * lane;
    v2f y = *(v2f*)yp;
    y.x += fsign(y.x) * nv.x * scale;
    y.y += fsign(y.y) * nv.y * scale;
    *(v2f*)yp = y;
}

// -------------------------------------------------------------------------
// Host-side orchestration (all on `stream`, graph-capture safe).
// -------------------------------------------------------------------------
extern "C" void kernel_launch(void* const* d_in, const int* in_sizes, int n_in,
                              void* d_out, int out_size, void* d_ws, size_t ws_size,
                              hipStream_t stream) {
    const float* user  = (const float*)d_in[0];   // [60000,64]
    const float* item  = (const float*)d_in[1];   // [40000,64]
    const float* W     = (const float*)d_in[2];   // [64,64]
    const float* vals  = (const float*)d_in[3];   // [2M]
    const float* noise = (const float*)d_in[4];   // [2,100000,64]
    const int*   rows  = (const int*)d_in[5];     // [2M]
    const int*   cols  = (const int*)d_in[6];     // [2M]

    float* out     = (float*)d_out;
    float* conbase = out;                          // [N_NODES,128] flat
    float* binbase = out + N_NODES * OUT_ROW;      // [N_NODES,128] flat

    const int B = 256;
    const int zero_blocks  = (N_NODES * CON_DIM + B - 1) / B;
    const int spmm_blocks  = (N_EDGES * 64) / B;          // 128M/256
    const int hash_blocks  = (N_NODES / 16 + 7) / 8;      // 1 wave = 16-node tile
    const int noise_blocks = (N_NODES + 7) / 8;           // 1 wave = 1 node

    for (int l = 0; l < 2; ++l) {
        zero_slice_kernel<<<zero_blocks, B, 0, stream>>>(conbase, l);
        spmm_scatter_kernel<<<spmm_blocks, B, 0, stream>>>(rows, cols, vals,
                                                           user, item, conbase, l);
        hash_wmma_kernel<<<hash_blocks, B, 0, stream>>>(conbase, W, binbase, l);
        noise_update_kernel<<<noise_blocks, B, 0, stream>>>(noise, conbase, l);
    }
}